// GCN_pia4_44306882625591
// MI455X (gfx1250) — compile-verified
//
#include <hip/hip_runtime.h>
#include <hip/hip_bf16.h>

// ---------------------------------------------------------------------------
// GCN on MI455X (gfx1250): fp32 WMMA GEMM (A-fragment reuse across 4 N-tiles,
// transposed+padded W in LDS, b64 LDS fragment loads) + atomic SPMM.
// Layer: e = SPMM(row,col,w, X@W) + b ; h = relu(e). Final: log_softmax(e5).
// d_out layout (floats): [logp N*16][e1 N*64][e2 N*64][e3 N*64][e4 N*64][e5 N*16]
// ---------------------------------------------------------------------------

typedef __attribute__((ext_vector_type(2))) float v2f;
typedef __attribute__((ext_vector_type(8))) float v8f;

#define GCN_N 50000
#define GCN_E 800000

// out[idx] = b[idx & (ncols-1)]  (seed accumulator with bias)
__global__ void gcn_init_bias(float* __restrict__ out, const float* __restrict__ b,
                              long long total, int colShift) {
    long long idx = (long long)blockIdx.x * blockDim.x + threadIdx.x;
    if (idx >= total) return;
    int mask = (1 << colShift) - 1;
    out[idx] = b[idx & mask];
}

// Y[nrows x NCOLS] = X[nrows x K] @ W[K x NCOLS], via V_WMMA_F32_16X16X4_F32.
// One wave per 16-row stripe; it computes ALL NCOLS/16 column tiles so each
// A fragment (global load) is reused NT times. W is staged transposed in LDS
// with an even pad (K+2) so a B fragment is one aligned ds_load_b64 and both
// staging writes and fragment reads are bank-conflict-free.
template <int K, int NCOLS>
__global__ void gcn_gemm_wmma_f32(const float* __restrict__ X,
                                  const float* __restrict__ W,
                                  float* __restrict__ Y,
                                  int nrows) {
    constexpr int NT   = NCOLS / 16;   // column tiles per wave (4 or 1)
    constexpr int KPAD = K + 2;        // even pad: aligned b64, no bank conflicts
    __shared__ float sWT[NCOLS * KPAD];

    // stage W transposed: sWT[n][k] = W[k][n]
    for (int i = threadIdx.x; i < K * NCOLS; i += blockDim.x) {
        int k = i / NCOLS, n = i % NCOLS;
        sWT[n * KPAD + k] = W[i];
    }
    __syncthreads();

    const int lane  = threadIdx.x & 31;
    const int wave  = blockIdx.x * (blockDim.x >> 5) + (threadIdx.x >> 5);
    const int tileM = wave;                    // wave-uniform
    if (tileM * 16 >= nrows) return;           // wave-uniform: EXEC stays all-1s

    // A 16x4 f32 layout: lanes 0-15 -> M=lane, K={0,1}; lanes 16-31 -> K={2,3}
    const int m15   = lane & 15;
    const int khalf = (lane >> 4) << 1;        // 0 or 2
    const float* __restrict__ xrow = X + (long long)(tileM * 16 + m15) * K;
    __builtin_prefetch(xrow, 0, 0);            // global_prefetch_b8 of this wave's stripe

    v8f acc[NT];
#pragma unroll
    for (int t = 0; t < NT; ++t) acc[t] = (v8f)0.0f;

    for (int k = 0; k < K; k += 4) {
        v2f a = *(const v2f*)(xrow + k + khalf);              // one b64 global load
#pragma unroll
        for (int t = 0; t < NT; ++t) {
            // B 4x16: lane picks N = t*16 + m15, half picks K pair
            v2f b = *(const v2f*)(sWT + (t * 16 + m15) * KPAD + k + khalf); // ds_load_b64
            acc[t] = __builtin_amdgcn_wmma_f32_16x16x4_f32(
                /*neg_a=*/false, a, /*neg_b=*/false, b,
                /*c_mod=*/(short)0, acc[t], /*reuse_a=*/false, /*reuse_b=*/false);
        }
    }

    // C/D 16x16 f32 layout: VGPR r, lanes 0-15 -> M=r, N=lane; lanes 16-31 -> M=8+r
    const int mbase = tileM * 16 + ((lane >> 4) << 3);
#pragma unroll
    for (int t = 0; t < NT; ++t) {
        const int ncol = t * 16 + m15;
#pragma unroll
        for (int r = 0; r < 8; ++r)
            Y[(long long)(mbase + r) * NCOLS + ncol] = acc[t][r];
    }
}

// out[row[e]] += w[e] * Y[col[e]]  -- one thread per (edge, 4-feature group)
__global__ void gcn_spmm_scatter(const int* __restrict__ erow,
                                 const int* __restrict__ ecol,
                                 const float* __restrict__ ew,
                                 const float* __restrict__ Y,
                                 float* __restrict__ out,
                                 long long nedges, int colShift) {
    const int gShift = colShift - 2;                 // groups of 4 features
    long long idx = (long long)blockIdx.x * blockDim.x + threadIdx.x;
    long long total = nedges << gShift;
    if (idx >= total) return;
    long long e = idx >> gShift;
    int f0 = (int)(idx & ((1 << gShift) - 1)) << 2;
    int src = ecol[e], dst = erow[e];
    float w = ew[e];
    const float4 v = *(const float4*)(Y + ((long long)src << colShift) + f0);
    float* o = out + ((long long)dst << colShift) + f0;
    atomicAdd(o + 0, w * v.x);
    atomicAdd(o + 1, w * v.y);
    atomicAdd(o + 2, w * v.z);
    atomicAdd(o + 3, w * v.w);
}

__global__ void gcn_relu(const float* __restrict__ e, float* __restrict__ h,
                         long long total) {
    long long idx = (long long)blockIdx.x * blockDim.x + threadIdx.x;
    if (idx >= total) return;
    float v = e[idx];
    h[idx] = v > 0.0f ? v : 0.0f;
}

// Per-node log_softmax over 16 classes.
__global__ void gcn_log_softmax16(const float* __restrict__ e5,
                                  float* __restrict__ logp, int n) {
    int i = blockIdx.x * blockDim.x + threadIdx.x;
    if (i >= n) return;
    const float* r = e5 + (long long)i * 16;
    float m = r[0];
#pragma unroll
    for (int j = 1; j < 16; ++j) m = fmaxf(m, r[j]);
    float s = 0.0f;
#pragma unroll
    for (int j = 0; j < 16; ++j) s += __expf(r[j] - m);
    float lse = m + __logf(s);
    float* o = logp + (long long)i * 16;
#pragma unroll
    for (int j = 0; j < 16; ++j) o[j] = r[j] - lse;
}

// ---------------------------------------------------------------------------

static inline long long cdiv(long long a, long long b) { return (a + b - 1) / b; }

extern "C" void kernel_launch(void* const* d_in, const int* in_sizes, int n_in,
                              void* d_out, int out_size, void* d_ws, size_t ws_size,
                              hipStream_t stream) {
    const int N = GCN_N, E = GCN_E;
    (void)in_sizes; (void)n_in; (void)out_size; (void)ws_size;

    const float* x    = (const float*)d_in[0];
    const int*   erow = (const int*)d_in[1];
    const int*   ecol = (const int*)d_in[2];
    const float* ew   = (const float*)d_in[3];
    const float* Wl[5] = { (const float*)d_in[4], (const float*)d_in[6],
                           (const float*)d_in[8], (const float*)d_in[10],
                           (const float*)d_in[12] };
    const float* bl[5] = { (const float*)d_in[5], (const float*)d_in[7],
                           (const float*)d_in[9], (const float*)d_in[11],
                           (const float*)d_in[13] };

    float* out = (float*)d_out;
    float* logp = out;                              // N*16
    float* eout[5];
    eout[0] = out + (long long)N * 16;              // e1: N*64
    eout[1] = eout[0] + (long long)N * 64;          // e2
    eout[2] = eout[1] + (long long)N * 64;          // e3
    eout[3] = eout[2] + (long long)N * 64;          // e4
    eout[4] = eout[3] + (long long)N * 64;          // e5: N*16

    float* bufY = (float*)d_ws;                     // N*64 f32
    float* bufH = bufY + (long long)N * 64;         // N*64 f32

    const int BLK = 256;                            // 8 waves per block
    const int gemmBlocks = (int)cdiv(N / 16, BLK / 32);
    const int cSh[5] = { 6, 6, 6, 6, 4 };

    const float* in = x;
    for (int l = 0; l < 5; ++l) {
        const int ncols = 1 << cSh[l];
        const long long tot = (long long)N * ncols;

        // 1) seed e_l with bias
        gcn_init_bias<<<(int)cdiv(tot, BLK), BLK, 0, stream>>>(eout[l], bl[l], tot, cSh[l]);

        // 2) dense GEMM via fp32 WMMA (one wave = full 16-row output stripe)
        if (l == 0)
            gcn_gemm_wmma_f32<128, 64><<<gemmBlocks, BLK, 0, stream>>>(in, Wl[l], bufY, N);
        else if (l < 4)
            gcn_gemm_wmma_f32<64, 64><<<gemmBlocks, BLK, 0, stream>>>(in, Wl[l], bufY, N);
        else
            gcn_gemm_wmma_f32<64, 16><<<gemmBlocks, BLK, 0, stream>>>(in, Wl[l], bufY, N);

        // 3) sparse scatter-accumulate into e_l
        const long long sthreads = (long long)E * (ncols >> 2);
        gcn_spmm_scatter<<<(int)cdiv(sthreads, BLK), BLK, 0, stream>>>(
            erow, ecol, ew, bufY, eout[l], E, cSh[l]);

        // 4) relu into next-layer input (layers 1..4)
        if (l < 4) {
            gcn_relu<<<(int)cdiv(tot, BLK), BLK, 0, stream>>>(eout[l], bufH, tot);
            in = bufH;
        }
    }

    gcn_log_softmax16<<<(int)cdiv(N, BLK), BLK, 0, stream>>>(eout[4], logp, N);
}